// TimestepNorm_45638322487508
// MI455X (gfx1250) — compile-verified
//
#include <hip/hip_runtime.h>
#include <stdint.h>

#define BB 8
#define LL 4096
#define DD 1024
#define GG 128
#define DG 8                 // features per group
#define CC 64                // chunks along L
#define SS (LL / CC)         // 64 steps per chunk
#define TSTEPS 8             // steps per LDS tile (32 KB)
#define NTILES (SS / TSTEPS) // 8 tiles per chunk
#define XFERS ((TSTEPS * DD) / (128 * 4)) // 16 b128 async xfers per thread per tile
#define EPSV 1e-5f

// ---------------------------------------------------------------------------
// Async global -> LDS 16B copy (gfx1250): GLOBAL_LOAD_ASYNC_TO_LDS_B128.
// Operands: LDS byte offset (wave-relative, = low 32 bits of the generic
// shared-space address) in a VGPR, 64-bit global address in a VGPR pair.
// ---------------------------------------------------------------------------
__device__ __forceinline__ void async_ld16(const float* g, float* l) {
  asm volatile("global_load_async_to_lds_b128 %0, %1, off"
               :: "v"((uint32_t)(uintptr_t)l), "v"((uint64_t)(uintptr_t)g)
               : "memory");
}

#define WAIT_ASYNC(n) asm volatile("s_wait_asynccnt %0" :: "n"(n) : "memory")

__device__ __forceinline__ void issue_tile(const float* __restrict__ xg,
                                           float* __restrict__ buf, int tid) {
#pragma unroll
  for (int i = 0; i < XFERS; ++i) {
    int off = (i * 128 + tid) * 4; // float index, 16B granules
    async_ld16(xg + off, buf + off);
  }
}

// ---------------------------------------------------------------------------
// PASS 1: per-chunk Welford aggregates (cnt, mean, M2) -> ws
// PASS 2: replay scan from prefix state in ws, write y
// One block = one (batch, chunk); thread g in [0,128) owns group g.
// ---------------------------------------------------------------------------
template <int PASS>
__global__ void __launch_bounds__(128)
tsnorm_chunks(const float* __restrict__ x, const int* __restrict__ pmask,
              const float* __restrict__ weight, const float* __restrict__ bias,
              float* __restrict__ wsc, float* __restrict__ wsm,
              float* __restrict__ wsv, float* __restrict__ y) {
  __shared__ float tile[2][TSTEPS * DD]; // 2 x 32 KB double buffer
  __shared__ float smask[SS];

  const int tid = threadIdx.x;
  const int bc = blockIdx.x;
  const int b = bc / CC;
  const int c = bc % CC;
  const int t0 = c * SS;
  const int g = tid;
  const int chain = bc * GG + g;

  if (tid < SS) smask[tid] = (float)pmask[b * LL + t0 + tid];

  float cnt, mean, M2;
  if (PASS == 2) {
    cnt = wsc[chain]; mean = wsm[chain]; M2 = wsv[chain];
  } else {
    cnt = 0.0f; mean = 0.0f; M2 = 0.0f;
  }

  float gam[DG], bet[DG];
  if (PASS == 2) {
#pragma unroll
    for (int j = 0; j < DG; ++j) {
      gam[j] = weight[g * DG + j] + 1.0f;
      bet[j] = bias[g * DG + j];
    }
  }

  const float* xg = x + ((size_t)b * LL + t0) * DD;
  float* yg = y + ((size_t)b * LL + t0) * DD;

  issue_tile(xg, tile[0], tid);

  for (int k = 0; k < NTILES; ++k) {
    if (k + 1 < NTILES) {
      issue_tile(xg + (size_t)(k + 1) * TSTEPS * DD, tile[(k + 1) & 1], tid);
      WAIT_ASYNC(XFERS); // previous tile done; only just-issued may be pending
    } else {
      WAIT_ASYNC(0);
    }
    __syncthreads();

    const float* tb = tile[k & 1];
#pragma unroll
    for (int s = 0; s < TSTEPS; ++s) {
      float xv[DG];
      const float* row = tb + s * DD + g * DG;
      *(float4*)&xv[0] = *(const float4*)&row[0];
      *(float4*)&xv[4] = *(const float4*)&row[4];

      float mu = 0.0f;
#pragma unroll
      for (int j = 0; j < DG; ++j) mu += xv[j];
      mu *= (1.0f / (float)DG);
      float ssq = 0.0f;
#pragma unroll
      for (int j = 0; j < DG; ++j) { float t = xv[j] - mu; ssq += t * t; }

      float N = cnt * (float)DG;
      float Np = N + (float)DG;            // >= 8
      float rNp = 1.0f / Np;
      float delta = mu - mean;
      float mean_n = mean + delta * ((float)DG * rNp);
      float M2_n = M2 + ssq + delta * delta * (N * (float)DG * rNp);
      float v = smask[k * TSTEPS + s];
      mean += v * (mean_n - mean);
      M2   += v * (M2_n - M2);
      cnt  += v;

      if (PASS == 2) {
        float Nu = fmaxf(cnt * (float)DG, (float)DG);
        float rstd = rsqrtf(M2 / Nu + EPSV);
        float out[DG];
#pragma unroll
        for (int j = 0; j < DG; ++j)
          out[j] = (xv[j] - mean) * rstd * gam[j] + bet[j];
        float* yrow = yg + (size_t)(k * TSTEPS + s) * DD + g * DG;
        *(float4*)&yrow[0] = *(float4*)&out[0];
        *(float4*)&yrow[4] = *(float4*)&out[4];
      }
    }
    __syncthreads(); // buffer may be overwritten next iteration
  }

  if (PASS == 1) { wsc[chain] = cnt; wsm[chain] = mean; wsv[chain] = M2; }
}

// ---------------------------------------------------------------------------
// Sequential Chan-merge over the C chunk aggregates per (b,g).
// Leaves exclusive-prefix state in ws; writes final count/mean/var.
// ---------------------------------------------------------------------------
__global__ void tsnorm_scan(const int* __restrict__ prev_count,
                            const float* __restrict__ prev_mean,
                            const float* __restrict__ prev_var,
                            float* __restrict__ wsc, float* __restrict__ wsm,
                            float* __restrict__ wsv,
                            float* __restrict__ count_out,
                            float* __restrict__ mean_out,
                            float* __restrict__ var_out) {
  int id = blockIdx.x * blockDim.x + threadIdx.x;
  if (id >= BB * GG) return;
  int b = id / GG, g = id % GG;

  float cnt = (float)prev_count[b];
  float mean = prev_mean[id];
  float M2 = prev_var[id] * (cnt * (float)DG);

  for (int c = 0; c < CC; ++c) {
    int idx = (b * CC + c) * GG + g;
    float acnt = wsc[idx], amean = wsm[idx], aM2 = wsv[idx];
    wsc[idx] = cnt; wsm[idx] = mean; wsv[idx] = M2; // exclusive prefix
    float nB = acnt * (float)DG;
    if (nB > 0.0f) {
      float nA = cnt * (float)DG;
      float n = nA + nB;
      float rn = 1.0f / n;
      float delta = amean - mean;
      mean += delta * nB * rn;
      M2 += aM2 + delta * delta * nA * nB * rn;
      cnt += acnt;
    }
  }
  mean_out[id] = mean;
  float Nf = cnt * (float)DG;
  var_out[id] = (Nf > 0.0f) ? (M2 / Nf) : prev_var[id];
  if (g == 0) count_out[b] = cnt;
}

// ---------------------------------------------------------------------------
extern "C" void kernel_launch(void* const* d_in, const int* in_sizes, int n_in,
                              void* d_out, int out_size, void* d_ws,
                              size_t ws_size, hipStream_t stream) {
  const float* x          = (const float*)d_in[0];
  const int*   prev_count = (const int*)d_in[1];
  const float* prev_mean  = (const float*)d_in[2];
  const float* prev_var   = (const float*)d_in[3];
  const float* weight     = (const float*)d_in[4];
  const float* bias       = (const float*)d_in[5];
  const int*   pmask      = (const int*)d_in[6];

  // outputs concatenated flat in return order: y, count_f, mean_f, var_f
  float* y         = (float*)d_out;
  float* count_out = y + (size_t)BB * LL * DD;
  float* mean_out  = count_out + BB;
  float* var_out   = mean_out + (size_t)BB * GG;

  float* wsc = (float*)d_ws;                 // B*G*C each
  float* wsm = wsc + (size_t)BB * GG * CC;
  float* wsv = wsm + (size_t)BB * GG * CC;

  dim3 grid(BB * CC), block(128);
  tsnorm_chunks<1><<<grid, block, 0, stream>>>(x, pmask, weight, bias,
                                               wsc, wsm, wsv, y);
  tsnorm_scan<<<(BB * GG + 255) / 256, 256, 0, stream>>>(
      prev_count, prev_mean, prev_var, wsc, wsm, wsv,
      count_out, mean_out, var_out);
  tsnorm_chunks<2><<<grid, block, 0, stream>>>(x, pmask, weight, bias,
                                               wsc, wsm, wsv, y);
}